// MessagePassing_v1_13099650253151
// MI455X (gfx1250) — compile-verified
//
#include <hip/hip_runtime.h>

#define N_NODES 10000
#define N_EDGES 320000
#define NS 128
#define NTYPES 8
#define FC 64

typedef float v2f __attribute__((ext_vector_type(2)));
typedef float v8f __attribute__((ext_vector_type(8)));

__device__ __forceinline__ float sigmoid_f(float x) {
    return 1.0f / (1.0f + expf(-x));
}
__device__ __forceinline__ float silu_f(float x) {
    return x * sigmoid_f(x);
}

// ---------------------------------------------------------------------------
// Per-(layer,type) weight table: w = silu(silu(f0[t]) @ f1) @ f2, scaled by
// 1/sqrt(32) (folded message normalization). grid = 3 layers * 8 types.
// ---------------------------------------------------------------------------
__global__ void __launch_bounds__(64)
wtab_kernel(const float* __restrict__ f0_0, const float* __restrict__ f1_0, const float* __restrict__ f2_0,
            const float* __restrict__ f0_1, const float* __restrict__ f1_1, const float* __restrict__ f2_1,
            const float* __restrict__ f0_2, const float* __restrict__ f1_2, const float* __restrict__ f2_2,
            float* __restrict__ wtab) {
    const int l = blockIdx.x >> 3;
    const int t = blockIdx.x & 7;
    const float* f0 = (l == 0) ? f0_0 : (l == 1) ? f0_1 : f0_2;
    const float* f1 = (l == 0) ? f1_0 : (l == 1) ? f1_1 : f1_2;
    const float* f2 = (l == 0) ? f2_0 : (l == 1) ? f2_1 : f2_2;
    const int dout = (l == 2) ? 64 : 384;

    __shared__ float t0s[FC];
    __shared__ float t1s[FC];
    const int j = threadIdx.x;

    t0s[j] = silu_f(f0[t * FC + j]);
    __syncthreads();

    float acc = 0.0f;
    #pragma unroll 8
    for (int k = 0; k < FC; ++k) acc += t0s[k] * f1[k * FC + j];
    t1s[j] = silu_f(acc);
    __syncthreads();

    const float scale = 0.17677669529663687f;  // 1/sqrt(32)
    for (int c = j; c < dout; c += FC) {
        float s = 0.0f;
        #pragma unroll 8
        for (int k = 0; k < FC; ++k) s += t1s[k] * f2[k * dout + c];
        wtab[(l * NTYPES + t) * 384 + c] = s * scale;
    }
}

// ---------------------------------------------------------------------------
// Y[N x dout] = X[N x di] @ W[di x dout], f32 WMMA 16x16x4.
// One wave -> 16x32 output slab (2 accumulators, shared A fetch).
// 8 waves per block -> 128 rows. N = 10000 = 625 row tiles exactly.
// ---------------------------------------------------------------------------
__global__ void __launch_bounds__(256)
gemm_wmma_kernel(const float* __restrict__ X, const float* __restrict__ W,
                 float* __restrict__ Y, int di, int dout) {
    const int lane = threadIdx.x & 31;
    const int wave = threadIdx.x >> 5;
    const int tile_m = blockIdx.x * 8 + wave;
    if (tile_m * 16 >= N_NODES) return;  // wave-uniform: EXEC stays all-1s

    const int li = lane & 15;
    const int hi = lane >> 4;
    const int m0 = tile_m * 16;
    const int n0 = blockIdx.y * 32;

    const float* xr  = X + (m0 + li) * di;       // A: row m0+li
    const float* wc0 = W + n0 + li;              // B cols for tile 0
    const float* wc1 = wc0 + 16;                 // B cols for tile 1

    v8f acc0 = {0.f, 0.f, 0.f, 0.f, 0.f, 0.f, 0.f, 0.f};
    v8f acc1 = {0.f, 0.f, 0.f, 0.f, 0.f, 0.f, 0.f, 0.f};

    #pragma unroll 4
    for (int k0 = 0; k0 < di; k0 += 4) {
        const int ka = k0 + 2 * hi;              // K pair owned by this lane half
        v2f a = *(const v2f*)(xr + ka);          // A: V0=K(ka), V1=K(ka+1)
        v2f b0, b1;
        b0.x = wc0[ka * dout];                   // B: V0=K(ka) at col n0+li
        b0.y = wc0[(ka + 1) * dout];             //    V1=K(ka+1)
        b1.x = wc1[ka * dout];
        b1.y = wc1[(ka + 1) * dout];
        acc0 = __builtin_amdgcn_wmma_f32_16x16x4_f32(false, a, false, b0,
                                                     (short)0, acc0, false, false);
        acc1 = __builtin_amdgcn_wmma_f32_16x16x4_f32(false, a, false, b1,
                                                     (short)0, acc1, false, false);
    }

    // C/D layout: VGPR i holds M = i + 8*hi, N = li (tile0) / li+16 (tile1)
    float* yp = Y + (m0 + hi * 8) * dout + n0 + li;
    #pragma unroll
    for (int i = 0; i < 8; ++i) {
        yp[i * dout]      = acc0[i];
        yp[i * dout + 16] = acc1[i];
    }
}

// ---------------------------------------------------------------------------
// h[dst] += y_msg[src] * w_table[type]   (scale already folded into table)
// One block per edge; blockDim == dout so each thread owns one column.
// All buffers are L2-resident (15 MB each vs 192 MB L2).
// ---------------------------------------------------------------------------
__global__ void scatter_kernel(const float* __restrict__ ymsg,
                               const float* __restrict__ wtab,
                               const int* __restrict__ esrc,
                               const int* __restrict__ edst,
                               const int* __restrict__ etyp,
                               float* __restrict__ hbuf, int dout) {
    const int e = blockIdx.x;
    const int src = esrc[e];
    const int dst = edst[e];
    const int ty  = etyp[e];
    const float* ym = ymsg + src * dout;
    const float* w  = wtab + ty * 384;
    float* hb = hbuf + dst * dout;
    for (int c = threadIdx.x; c < dout; c += blockDim.x) {
        atomicAdd(&hb[c], ym[c] * w[c]);
    }
}

// ---------------------------------------------------------------------------
// mid = concat(node_pos[64], silu(h[:,0:128]), h[:,256:384]*sigmoid(h[:,128:256]))
// One block per node, 320 threads = one thread per output column.
// ---------------------------------------------------------------------------
__global__ void __launch_bounds__(320)
gate_kernel(const float* __restrict__ h, const float* __restrict__ node_pos,
            float* __restrict__ xout) {
    const int n = blockIdx.x;
    const int j = threadIdx.x;
    const float* hn = h + n * 384;
    float v;
    if (j < 64) {
        v = node_pos[n * 64 + j];
    } else if (j < 64 + NS) {
        v = silu_f(hn[j - 64]);
    } else {
        const int c = j - (64 + NS);
        v = hn[256 + c] * sigmoid_f(hn[128 + c]);
    }
    xout[n * 320 + j] = v;
}

// ---------------------------------------------------------------------------
extern "C" void kernel_launch(void* const* d_in, const int* in_sizes, int n_in,
                              void* d_out, int out_size, void* d_ws, size_t ws_size,
                              hipStream_t stream) {
    (void)in_sizes; (void)n_in; (void)out_size; (void)ws_size;

    const float* node_pos = (const float*)d_in[0];
    const float* w_self[3] = {(const float*)d_in[1], (const float*)d_in[6],  (const float*)d_in[11]};
    const float* w_msg[3]  = {(const float*)d_in[2], (const float*)d_in[7],  (const float*)d_in[12]};
    const float* fc0[3]    = {(const float*)d_in[3], (const float*)d_in[8],  (const float*)d_in[13]};
    const float* fc1[3]    = {(const float*)d_in[4], (const float*)d_in[9],  (const float*)d_in[14]};
    const float* fc2[3]    = {(const float*)d_in[5], (const float*)d_in[10], (const float*)d_in[15]};
    const int* esrc = (const int*)d_in[16];
    const int* edst = (const int*)d_in[17];
    const int* etyp = (const int*)d_in[18];
    float* out = (float*)d_out;

    // Workspace carve-up (all f32): 9216 + 3.2M + 3.84M + 3.84M floats ~= 44 MB
    float* wtab = (float*)d_ws;                          // [3][8][384]
    float* xbuf = wtab + 3 * NTYPES * 384;               // [N][320]
    float* hbuf = xbuf + (size_t)N_NODES * 320;          // [N][384]
    float* ymsg = hbuf + (size_t)N_NODES * 384;          // [N][384]

    const dim3 gemm_blk(256);
    const dim3 gemm_grid_384(79, 12);  // ceil(625/8) x (384/32)
    const dim3 gemm_grid_64(79, 2);    // ceil(625/8) x (64/32)

    // Edge-type MLP tables (tiny, once).
    wtab_kernel<<<24, 64, 0, stream>>>(fc0[0], fc1[0], fc2[0],
                                       fc0[1], fc1[1], fc2[1],
                                       fc0[2], fc1[2], fc2[2], wtab);

    // ---- Layer 0: x = node_pos (di=64, do=384) ----
    gemm_wmma_kernel<<<gemm_grid_384, gemm_blk, 0, stream>>>(node_pos, w_self[0], hbuf, 64, 384);
    gemm_wmma_kernel<<<gemm_grid_384, gemm_blk, 0, stream>>>(node_pos, w_msg[0],  ymsg, 64, 384);
    scatter_kernel<<<N_EDGES, 384, 0, stream>>>(ymsg, wtab, esrc, edst, etyp, hbuf, 384);
    gate_kernel<<<N_NODES, 320, 0, stream>>>(hbuf, node_pos, xbuf);

    // ---- Layer 1: x = xbuf (di=320, do=384) ----
    gemm_wmma_kernel<<<gemm_grid_384, gemm_blk, 0, stream>>>(xbuf, w_self[1], hbuf, 320, 384);
    gemm_wmma_kernel<<<gemm_grid_384, gemm_blk, 0, stream>>>(xbuf, w_msg[1],  ymsg, 320, 384);
    scatter_kernel<<<N_EDGES, 384, 0, stream>>>(ymsg, wtab + NTYPES * 384, esrc, edst, etyp, hbuf, 384);
    gate_kernel<<<N_NODES, 320, 0, stream>>>(hbuf, node_pos, xbuf);

    // ---- Layer 2: x = xbuf (di=320, do=64), result straight into d_out ----
    gemm_wmma_kernel<<<gemm_grid_64, gemm_blk, 0, stream>>>(xbuf, w_self[2], out,  320, 64);
    gemm_wmma_kernel<<<gemm_grid_64, gemm_blk, 0, stream>>>(xbuf, w_msg[2],  ymsg, 320, 64);
    scatter_kernel<<<N_EDGES, 64, 0, stream>>>(ymsg, wtab + 2 * NTYPES * 384, esrc, edst, etyp, out, 64);
}